// HiddenMarkov_4672924418268
// MI455X (gfx1250) — compile-verified
//
#include <hip/hip_runtime.h>
#include <math.h>

typedef __attribute__((ext_vector_type(2))) float v2f;
typedef __attribute__((ext_vector_type(8))) float v8f;

#define T_LEN 2048
#define S_ST  128
#define M_MIX 8
#define D_DIM 64
#define KTOT  128   // 2*D  (concatenated [X^2 | X] / [iv | -2*mu*iv])
#define SMTOT 1024  // S*M

// ---------------------------------------------------------------------------
// Kernel 1: pack A = [X^2, X]  (T x 128), row-major
// ---------------------------------------------------------------------------
__global__ void prep_A(const float* __restrict__ X, float* __restrict__ Ap) {
  int idx = blockIdx.x * blockDim.x + threadIdx.x;   // T*D threads
  int t = idx >> 6, d = idx & 63;
  float x = X[idx];
  Ap[t * KTOT + d]         = x * x;
  Ap[t * KTOT + D_DIM + d] = x;
}

// ---------------------------------------------------------------------------
// Kernel 2: pack Bt[sm] = [iv, -2*mu*iv] (1024 x 128) and per-column constant
//   C[sm] = -0.5*(sum mu^2*iv + sum log_var + D*log(2pi)) + log_softmax(w)[s,m]
// ---------------------------------------------------------------------------
__global__ void prep_B(const float* __restrict__ means,
                       const float* __restrict__ log_vars,
                       const float* __restrict__ weight_logits,
                       float* __restrict__ Bt, float* __restrict__ Cc) {
  int sm = blockIdx.x;     // 0..1023
  int d  = threadIdx.x;    // 0..63
  float lv = log_vars[sm * D_DIM + d];
  float iv = __expf(-lv);
  float mu = means[sm * D_DIM + d];
  Bt[sm * KTOT + d]         = iv;
  Bt[sm * KTOT + D_DIM + d] = -2.0f * mu * iv;

  __shared__ float sq[64], sl[64];
  sq[d] = mu * mu * iv;
  sl[d] = lv;
  __syncthreads();
  for (int off = 32; off > 0; off >>= 1) {
    if (d < off) { sq[d] += sq[d + off]; sl[d] += sl[d + off]; }
    __syncthreads();
  }
  if (d == 0) {
    int s = sm >> 3, m = sm & 7;
    const float* w = weight_logits + s * M_MIX;
    float mx = w[0];
    #pragma unroll
    for (int i = 1; i < M_MIX; ++i) mx = fmaxf(mx, w[i]);
    float sum = 0.f;
    #pragma unroll
    for (int i = 0; i < M_MIX; ++i) sum += __expf(w[i] - mx);
    float lw = w[m] - (mx + __logf(sum));
    const float LOG2PI = 1.8378770664093453f;
    Cc[sm] = -0.5f * (sq[0] + sl[0] + (float)D_DIM * LOG2PI) + lw;
  }
}

// ---------------------------------------------------------------------------
// Kernel 3: FP32 WMMA GEMM (G = A * Bt^T) fused with mixture logsumexp.
// One wave per 16(time) x 16(sm) tile; K=128 via 32x v_wmma_f32_16x16x4_f32.
// A frag (ISA 16x4): lane<16 -> row lane&15, K=k..k+1 ; lane>=16 -> K=k+2..k+3
// B frag (4x16):     lane's N = lane&15, same K split -> float2 from Bt row.
// C/D: c[v] = G[t0 + v + 8*(lane>>4)][n0 + (lane&15)]
// ---------------------------------------------------------------------------
__global__ __launch_bounds__(256) void gemm_emit(
    const float* __restrict__ Ap, const float* __restrict__ Bt,
    const float* __restrict__ Cc, float* __restrict__ em) {
  int wid  = (blockIdx.x * blockDim.x + threadIdx.x) >> 5;
  int lane = threadIdx.x & 31;
  int tileN = wid & 63;                  // 64 tiles along SM
  int tileT = wid >> 6;                  // 128 tiles along T
  int t0 = tileT * 16, n0 = tileN * 16;
  int r  = lane & 15;
  int kh = (lane >> 4) << 1;             // 0 or 2
  const float* Arow = Ap + (size_t)(t0 + r) * KTOT + kh;
  const float* Brow = Bt + (size_t)(n0 + r) * KTOT + kh;

  v8f c = {};
  #pragma unroll
  for (int k = 0; k < KTOT; k += 4) {
    v2f a = *(const v2f*)(Arow + k);
    v2f b = *(const v2f*)(Brow + k);
    c = __builtin_amdgcn_wmma_f32_16x16x4_f32(
        /*neg_a=*/false, a, /*neg_b=*/false, b,
        /*c_mod=*/(short)0, c, /*reuse_a=*/false, /*reuse_b=*/false);
  }

  float cc   = Cc[n0 + r];                       // this lane's column constant
  int state  = (n0 >> 3) + ((lane >> 3) & 1);    // 2 states per tile
  int tbase  = t0 + ((lane >> 4) << 3);          // +8 rows for upper half-wave

  #pragma unroll
  for (int v = 0; v < 8; ++v) {
    float val = -0.5f * c[v] + cc;               // comp_log_prob + log_mix_w
    // logsumexp across the 8 mixture lanes (xor butterfly, width 8)
    float mx = val;
    #pragma unroll
    for (int off = 1; off < 8; off <<= 1)
      mx = fmaxf(mx, __shfl_xor(mx, off, 8));
    float se = __expf(val - mx);
    #pragma unroll
    for (int off = 1; off < 8; off <<= 1)
      se += __shfl_xor(se, off, 8);
    float e = mx + __logf(se);
    if ((lane & 7) == 0)
      em[(size_t)(tbase + v) * S_ST + state] = e;
  }
}

// ---------------------------------------------------------------------------
// Kernel 4: finalize.
// alpha_last[s] = log_pi[s] + sum_t em[t,s] + (T-1)*rowLSE(log_A)[s]
// out = logsumexp_s(alpha_last)
// ---------------------------------------------------------------------------
__global__ __launch_bounds__(128) void finalize(
    const float* __restrict__ em, const float* __restrict__ pi_logits,
    const float* __restrict__ trans, float* __restrict__ out) {
  int s = threadIdx.x;  // 0..127

  float acc = 0.f;
  for (int t = 0; t < T_LEN; ++t) acc += em[(size_t)t * S_ST + s];

  const float* row = trans + (size_t)s * S_ST;
  float mx = -INFINITY;
  for (int j = 0; j < S_ST; ++j) mx = fmaxf(mx, row[j]);
  float sum = 0.f;
  for (int j = 0; j < S_ST; ++j) sum += __expf(row[j] - mx);
  float lse1 = mx + __logf(sum);                 // row LSE of trans_logits
  float mx2  = mx - lse1;                        // max of log_A row
  float sum2 = 0.f;
  for (int j = 0; j < S_ST; ++j) sum2 += __expf(row[j] - lse1 - mx2);
  float rlse = mx2 + __logf(sum2);

  __shared__ float sh[S_ST];
  float p = pi_logits[s];
  sh[s] = p; __syncthreads();
  for (int off = 64; off > 0; off >>= 1) {
    if (s < off) sh[s] = fmaxf(sh[s], sh[s + off]);
    __syncthreads();
  }
  float pmax = sh[0]; __syncthreads();
  sh[s] = __expf(p - pmax); __syncthreads();
  for (int off = 64; off > 0; off >>= 1) {
    if (s < off) sh[s] += sh[s + off];
    __syncthreads();
  }
  float plse = pmax + __logf(sh[0]); __syncthreads();

  float alpha = (p - plse) + acc + (float)(T_LEN - 1) * rlse;

  sh[s] = alpha; __syncthreads();
  for (int off = 64; off > 0; off >>= 1) {
    if (s < off) sh[s] = fmaxf(sh[s], sh[s + off]);
    __syncthreads();
  }
  float amax = sh[0]; __syncthreads();
  sh[s] = __expf(alpha - amax); __syncthreads();
  for (int off = 64; off > 0; off >>= 1) {
    if (s < off) sh[s] += sh[s + off];
    __syncthreads();
  }
  if (s == 0) out[0] = amax + __logf(sh[0]);
}

// ---------------------------------------------------------------------------
extern "C" void kernel_launch(void* const* d_in, const int* in_sizes, int n_in,
                              void* d_out, int out_size, void* d_ws, size_t ws_size,
                              hipStream_t stream) {
  (void)in_sizes; (void)n_in; (void)out_size; (void)ws_size;
  const float* X  = (const float*)d_in[0];
  const float* pi = (const float*)d_in[1];
  const float* tr = (const float*)d_in[2];
  const float* wl = (const float*)d_in[3];
  const float* mu = (const float*)d_in[4];
  const float* lv = (const float*)d_in[5];

  char*  ws = (char*)d_ws;
  float* Ap = (float*)(ws);                         // 2048*128*4 = 1 MiB
  float* Bt = (float*)(ws + 1048576);               // 1024*128*4 = 512 KiB
  float* Cc = (float*)(ws + 1048576 + 524288);      // 1024*4     = 4 KiB
  float* em = (float*)(ws + 1048576 + 524288 + 4096); // 2048*128*4 = 1 MiB

  hipLaunchKernelGGL(prep_A, dim3((T_LEN * D_DIM) / 256), dim3(256), 0, stream,
                     X, Ap);
  hipLaunchKernelGGL(prep_B, dim3(SMTOT), dim3(64), 0, stream,
                     mu, lv, wl, Bt, Cc);
  // 128*64 = 8192 tiles, 8 waves (256 threads) per block -> 1024 blocks
  hipLaunchKernelGGL(gemm_emit, dim3(1024), dim3(256), 0, stream,
                     Ap, Bt, Cc, em);
  hipLaunchKernelGGL(finalize, dim3(1), dim3(128), 0, stream,
                     em, pi, tr, (float*)d_out);
}